// MMDNE_51410758533802
// MI455X (gfx1250) — compile-verified
//
#include <hip/hip_runtime.h>
#include <hip/hip_bf16.h>

typedef float v2f __attribute__((ext_vector_type(2)));
typedef float v8f __attribute__((ext_vector_type(8)));

#define BB 4096
#define HH 10
#define NN 10
#define FF 64
#define EE 128
#define GG 32
#define NEG_SLOPE 0.2f

#define R1 (42 * BB)   // total gathered embedding rows
#define R2 (22 * BB)   // rows projected through W (s, t, s_h, t_h)

// workspace layout (bytes)
#define IDX_BYTES ((size_t)R1 * 4)
#define EMB_BYTES ((size_t)R1 * EE * 4)
#define HW_OFFSET (IDX_BYTES + EMB_BYTES)

// ---------------------------------------------------------------------------
// Kernel 0: concatenate all gather indices into one flat list.
// Row order: s(B) | t(B) | s_h(10B) | t_h(10B) | s_neg(10B) | t_neg(10B)
// ---------------------------------------------------------------------------
__global__ __launch_bounds__(256) void build_idx_kernel(
    const int* __restrict__ s_nodes, const int* __restrict__ t_nodes,
    const int* __restrict__ s_h, const int* __restrict__ t_h,
    const int* __restrict__ s_neg, const int* __restrict__ t_neg,
    int* __restrict__ idx)
{
    int r = blockIdx.x * 256 + threadIdx.x;
    if (r >= R1) return;
    int v;
    if      (r <  BB)      v = s_nodes[r];
    else if (r < 2*BB)     v = t_nodes[r - BB];
    else if (r < 12*BB)    v = s_h[r - 2*BB];
    else if (r < 22*BB)    v = t_h[r - 12*BB];
    else if (r < 32*BB)    v = s_neg[r - 22*BB];
    else                   v = t_neg[r - 32*BB];
    idx[r] = v;
}

// ---------------------------------------------------------------------------
// Kernel 1: embedding GEMM. emb[r, :] = user_fts[idx[r], :] @ f2e_w + f2e_b
// (R1 x 64) x (64 x 128).  16-row tile per block, 8 waves, each wave owns a
// 16-column N-tile.  K loop = 16 steps of V_WMMA_F32_16X16X4_F32.
// ---------------------------------------------------------------------------
__global__ __launch_bounds__(256) void emb_gemm_kernel(
    const float* __restrict__ user_fts, const float* __restrict__ wmat,
    const float* __restrict__ bias, const int* __restrict__ idx,
    float* __restrict__ emb)
{
    __shared__ float As[16][FF + 4];   // padded to dodge bank conflicts
    const int t    = threadIdx.x;
    const int tile = blockIdx.x;

    // stage gathered 16x64 A tile into LDS (each thread: 4 floats)
    {
        int r  = t >> 4;            // 0..15
        int c0 = (t & 15) * 4;      // 0..60
        int g  = idx[tile * 16 + r];
        const float* src = user_fts + (long)g * FF + c0;
        As[r][c0 + 0] = src[0];
        As[r][c0 + 1] = src[1];
        As[r][c0 + 2] = src[2];
        As[r][c0 + 3] = src[3];
    }
    __syncthreads();

    const int wave = t >> 5;
    const int lane = t & 31;
    const int m    = lane & 15;     // row within tile (A) / col within N-tile (B,C)
    const int kh   = lane >> 4;     // K-half selector
    const int n    = (wave << 4) + m;   // output column 0..127

    v8f c = {0.f, 0.f, 0.f, 0.f, 0.f, 0.f, 0.f, 0.f};
#pragma unroll
    for (int ks = 0; ks < 16; ++ks) {
        const int kk = ks * 4 + kh * 2;
        v2f a, b;
        a.x = As[m][kk];
        a.y = As[m][kk + 1];
        b.x = wmat[(kk    ) * EE + n];
        b.y = wmat[(kk + 1) * EE + n];
        c = __builtin_amdgcn_wmma_f32_16x16x4_f32(false, a, false, b,
                                                  (short)0, c, false, false);
    }

    const float bn   = bias[n];
    const long  base = (long)tile * 16;
#pragma unroll
    for (int r = 0; r < 8; ++r) {
        const long row = base + r + kh * 8;    // C/D layout: VGPR r -> M=r (+8 upper half)
        emb[row * EE + n] = c[r] + bn;
    }
}

// ---------------------------------------------------------------------------
// Kernel 2: projection GEMM. hw[r, :] = emb[r, :] @ W  for r in [0, 22B)
// (R2 x 128) x (128 x 32).  16-row tile per block, 2 waves (one N-tile each),
// K loop = 32 WMMA steps.
// ---------------------------------------------------------------------------
__global__ __launch_bounds__(64) void hw_gemm_kernel(
    const float* __restrict__ emb, const float* __restrict__ wmat,
    float* __restrict__ hw)
{
    __shared__ float As[16][EE + 4];
    const int t    = threadIdx.x;
    const int tile = blockIdx.x;

    // stage 16x128 A tile (each of 64 threads: 32 floats)
    {
        int r  = t >> 2;            // 0..15
        int c0 = (t & 3) * 32;      // 0,32,64,96
        const float* src = emb + ((long)tile * 16 + r) * EE + c0;
#pragma unroll
        for (int j = 0; j < 32; ++j) As[r][c0 + j] = src[j];
    }
    __syncthreads();

    const int wave = t >> 5;
    const int lane = t & 31;
    const int m    = lane & 15;
    const int kh   = lane >> 4;
    const int n    = wave * 16 + m;   // 0..31

    v8f c = {0.f, 0.f, 0.f, 0.f, 0.f, 0.f, 0.f, 0.f};
#pragma unroll
    for (int ks = 0; ks < 32; ++ks) {
        const int kk = ks * 4 + kh * 2;
        v2f a, b;
        a.x = As[m][kk];
        a.y = As[m][kk + 1];
        b.x = wmat[(kk    ) * GG + n];
        b.y = wmat[(kk + 1) * GG + n];
        c = __builtin_amdgcn_wmma_f32_16x16x4_f32(false, a, false, b,
                                                  (short)0, c, false, false);
    }

    const long base = (long)tile * 16;
#pragma unroll
    for (int r = 0; r < 8; ++r) {
        hw[(base + r + kh * 8) * GG + n] = c[r];
    }
}

// ---------------------------------------------------------------------------
// Kernel 3: fused per-batch-element attention / gate / distance scoring.
// One block (256 threads) per batch element; everything staged in LDS.
// ---------------------------------------------------------------------------
__global__ __launch_bounds__(256) void fuse_kernel(
    const float* __restrict__ emb, const float* __restrict__ hw,
    const float* __restrict__ e_times,
    const float* __restrict__ s_h_times, const float* __restrict__ t_h_times,
    const float* __restrict__ s_mask,    const float* __restrict__ t_mask,
    const float* __restrict__ avec,      const float* __restrict__ ga_w,
    const float* __restrict__ ga_b,      const float* __restrict__ delta_s,
    const float* __restrict__ max_d,     float* __restrict__ out)
{
    __shared__ float se[EE], te[EE];
    __shared__ float sh[HH * EE], th[HH * EE], sn[NN * EE], tn[NN * EE];
    __shared__ float nWs[GG], nWt[GG], hWs[HH * GG], hWt[HH * GG];
    __shared__ float att_s[HH], att_t[HH];          // softmax weights
    __shared__ float aw_s[HH], aw_t[HH];            // att * mask
    __shared__ float w_s[HH], w_t[HH];              // att * exp(ds*d) * mask
    __shared__ float sim_s[HH], sim_t[HH], dts[HH], dtt[HH];
    __shared__ float pa_s[HH], pa_t[HH], nmu_s[NN], nmu_t[NN];
    __shared__ float na_s[HH * NN], na_t[HH * NN];
    __shared__ float shat[GG], that_[GG];
    __shared__ float gs_gt[2], dec_st[2], pmu_sh;

    const int b = blockIdx.x;
    const int t = threadIdx.x;
    const float md = max_d[0];
    const float ds = delta_s[0];

    // ---- cooperative LDS staging -------------------------------------
    const float* se_g = emb + (long)b * EE;
    const float* te_g = emb + (long)(BB + b) * EE;
    const float* sh_g = emb + (long)(2 * BB + b * HH) * EE;
    const float* th_g = emb + (long)(12 * BB + b * HH) * EE;
    const float* sn_g = emb + (long)(22 * BB + b * NN) * EE;
    const float* tn_g = emb + (long)(32 * BB + b * NN) * EE;
    for (int i = t; i < EE; i += 256)      { se[i] = se_g[i]; te[i] = te_g[i]; }
    for (int i = t; i < HH * EE; i += 256) { sh[i] = sh_g[i]; th[i] = th_g[i]; }
    for (int i = t; i < NN * EE; i += 256) { sn[i] = sn_g[i]; tn[i] = tn_g[i]; }
    const float* nws_g = hw + (long)b * GG;
    const float* nwt_g = hw + (long)(BB + b) * GG;
    const float* hws_g = hw + (long)(2 * BB + b * HH) * GG;
    const float* hwt_g = hw + (long)(12 * BB + b * HH) * GG;
    for (int i = t; i < GG; i += 256)      { nWs[i] = nws_g[i]; nWt[i] = nwt_g[i]; }
    for (int i = t; i < HH * GG; i += 256) { hWs[i] = hws_g[i]; hWt[i] = hwt_g[i]; }
    __syncthreads();

    // ---- per-history attention logits --------------------------------
    if (t < HH) {
        const int h = t;
        const float d = tanhf(fabsf(e_times[b] - s_h_times[b * HH + h]) / md);
        dts[h] = d;
        float s1 = 0.f, s2 = 0.f;
        for (int g = 0; g < GG; ++g) {
            s1 += nWs[g] * avec[g];
            s2 += hWs[h * GG + g] * avec[GG + g];
        }
        const float x = expf(-ds * d) * (s1 + s2);
        sim_s[h] = x > 0.f ? x : NEG_SLOPE * x;
    }
    if (t >= 32 && t < 32 + HH) {
        const int h = t - 32;
        const float d = tanhf(fabsf(e_times[b] - t_h_times[b * HH + h]) / md);
        dtt[h] = d;
        float s1 = 0.f, s2 = 0.f;
        for (int g = 0; g < GG; ++g) {
            s1 += nWt[g] * avec[g];
            s2 += hWt[h * GG + g] * avec[GG + g];
        }
        const float x = expf(-ds * d) * (s1 + s2);
        sim_t[h] = x > 0.f ? x : NEG_SLOPE * x;
    }
    __syncthreads();

    // ---- softmax over H, decay, weight vectors -----------------------
    if (t == 0) {
        float mx = -1e30f;
        for (int h = 0; h < HH; ++h) mx = fmaxf(mx, sim_s[h]);
        float ssum = 0.f, ex[HH];
        for (int h = 0; h < HH; ++h) { ex[h] = expf(sim_s[h] - mx); ssum += ex[h]; }
        float dmean = 0.f;
        for (int h = 0; h < HH; ++h) {
            const float at = ex[h] / ssum;
            const float mk = s_mask[b * HH + h];
            att_s[h] = at;
            aw_s[h]  = at * mk;
            w_s[h]   = at * expf(ds * dts[h]) * mk;
            dmean   += dts[h];
        }
        dec_st[0] = expf(-ds * dmean / (float)HH);
    }
    if (t == 32) {
        float mx = -1e30f;
        for (int h = 0; h < HH; ++h) mx = fmaxf(mx, sim_t[h]);
        float ssum = 0.f, ex[HH];
        for (int h = 0; h < HH; ++h) { ex[h] = expf(sim_t[h] - mx); ssum += ex[h]; }
        float dmean = 0.f;
        for (int h = 0; h < HH; ++h) {
            const float at = ex[h] / ssum;
            const float mk = t_mask[b * HH + h];
            att_t[h] = at;
            aw_t[h]  = at * mk;
            w_t[h]   = at * expf(ds * dtt[h]) * mk;
            dmean   += dtt[h];
        }
        dec_st[1] = expf(-ds * dmean / (float)HH);
    }
    __syncthreads();

    // ---- aggregated hidden states (scaled by decay, for the gate) ----
    if (t < GG) {
        float acc = 0.f;
        for (int h = 0; h < HH; ++h) acc += aw_s[h] * hWs[h * GG + t];
        shat[t] = acc * dec_st[0];
    }
    if (t >= 32 && t < 32 + GG) {
        const int g = t - 32;
        float acc = 0.f;
        for (int h = 0; h < HH; ++h) acc += aw_t[h] * hWt[h * GG + g];
        that_[g] = acc * dec_st[1];
    }
    __syncthreads();

    // ---- squared-distance task farm (241 tasks over 256 threads) -----
    if (t < 100) {
        const int h = t / 10, n = t % 10;
        float acc = 0.f;
        for (int e = 0; e < EE; ++e) {
            const float d = sh[h * EE + e] - tn[n * EE + e];
            acc += d * d;
        }
        na_s[h * NN + n] = -acc;
    } else if (t < 200) {
        const int q = t - 100, h = q / 10, n = q % 10;
        float acc = 0.f;
        for (int e = 0; e < EE; ++e) {
            const float d = th[h * EE + e] - sn[n * EE + e];
            acc += d * d;
        }
        na_t[h * NN + n] = -acc;
    } else if (t < 210) {
        const int h = t - 200;
        float acc = 0.f;
        for (int e = 0; e < EE; ++e) {
            const float d = sh[h * EE + e] - te[e];
            acc += d * d;
        }
        pa_s[h] = -acc;
    } else if (t < 220) {
        const int h = t - 210;
        float acc = 0.f;
        for (int e = 0; e < EE; ++e) {
            const float d = th[h * EE + e] - se[e];
            acc += d * d;
        }
        pa_t[h] = -acc;
    } else if (t < 230) {
        const int n = t - 220;
        float acc = 0.f;
        for (int e = 0; e < EE; ++e) {
            const float d = se[e] - tn[n * EE + e];
            acc += d * d;
        }
        nmu_s[n] = -acc;
    } else if (t < 240) {
        const int n = t - 230;
        float acc = 0.f;
        for (int e = 0; e < EE; ++e) {
            const float d = te[e] - sn[n * EE + e];
            acc += d * d;
        }
        nmu_t[n] = -acc;
    } else if (t == 240) {
        float acc = 0.f;
        for (int e = 0; e < EE; ++e) {
            const float d = se[e] - te[e];
            acc += d * d;
        }
        pmu_sh = -acc;
    }
    // gate (runs concurrently with the distance farm; thread 0's task above
    // is also executed — both writes are to distinct LDS locations)
    if (t == 0) {
        float zs = 0.f, zt = 0.f;
        for (int g = 0; g < GG; ++g) {
            zs += shat[g] * ga_w[g];
            zt += that_[g] * ga_w[g];
        }
        zs = tanhf(zs + ga_b[0]);
        zt = tanhf(zt + ga_b[0]);
        const float mx = fmaxf(zs, zt);
        const float es = expf(zs - mx), et = expf(zt - mx);
        gs_gt[0] = es / (es + et);
        gs_gt[1] = et / (es + et);
    }
    __syncthreads();

    // ---- final combine ------------------------------------------------
    const float gs = gs_gt[0], gt = gs_gt[1];
    if (t < NN) {
        float acc = 0.f;
        for (int h = 0; h < HH; ++h) acc += w_s[h] * na_s[h * NN + t];
        out[BB + b * NN + t] = nmu_s[t] + gs * acc;                 // n_lambda_s
    }
    if (t >= 32 && t < 32 + NN) {
        const int n = t - 32;
        float acc = 0.f;
        for (int h = 0; h < HH; ++h) acc += w_t[h] * na_t[h * NN + n];
        out[BB + BB * NN + b * NN + n] = nmu_t[n] + gt * acc;       // n_lambda_t
    }
    if (t == 64) {
        float accs = 0.f, acct = 0.f;
        for (int h = 0; h < HH; ++h) {
            accs += w_s[h] * pa_s[h];
            acct += w_t[h] * pa_t[h];
        }
        out[b] = pmu_sh + gs * accs + gt * acct;                    // p_lambda
    }
}

// ---------------------------------------------------------------------------
extern "C" void kernel_launch(void* const* d_in, const int* in_sizes, int n_in,
                              void* d_out, int out_size, void* d_ws, size_t ws_size,
                              hipStream_t stream)
{
    const int*   s_nodes   = (const int*)  d_in[0];
    const int*   t_nodes   = (const int*)  d_in[1];
    const float* e_times   = (const float*)d_in[2];
    const int*   s_h_nodes = (const int*)  d_in[3];
    const float* s_h_times = (const float*)d_in[4];
    const float* s_h_mask  = (const float*)d_in[5];
    const int*   t_h_nodes = (const int*)  d_in[6];
    const float* t_h_times = (const float*)d_in[7];
    const float* t_h_mask  = (const float*)d_in[8];
    const int*   s_neg     = (const int*)  d_in[9];
    const int*   t_neg     = (const int*)  d_in[10];
    const float* max_d     = (const float*)d_in[11];
    const float* user_fts  = (const float*)d_in[12];
    const float* f2e_w     = (const float*)d_in[13];
    const float* f2e_b     = (const float*)d_in[14];
    const float* Wm        = (const float*)d_in[15];
    const float* avec      = (const float*)d_in[16];
    const float* delta_s   = (const float*)d_in[17];
    const float* ga_w      = (const float*)d_in[18];
    const float* ga_b      = (const float*)d_in[19];
    float* out = (float*)d_out;

    char*  ws   = (char*)d_ws;
    int*   idx  = (int*)ws;
    float* embf = (float*)(ws + IDX_BYTES);
    float* hwf  = (float*)(ws + HW_OFFSET);

    build_idx_kernel<<<(R1 + 255) / 256, 256, 0, stream>>>(
        s_nodes, t_nodes, s_h_nodes, t_h_nodes, s_neg, t_neg, idx);

    emb_gemm_kernel<<<R1 / 16, 256, 0, stream>>>(user_fts, f2e_w, f2e_b, idx, embf);

    hw_gemm_kernel<<<R2 / 16, 64, 0, stream>>>(embf, Wm, hwf);

    fuse_kernel<<<BB, 256, 0, stream>>>(embf, hwf, e_times, s_h_times, t_h_times,
                                        s_h_mask, t_h_mask, avec, ga_w, ga_b,
                                        delta_s, max_d, out);
}